// QuantumLayer_44504451121459
// MI455X (gfx1250) — compile-verified
//
#include <hip/hip_runtime.h>

#define PI_F 3.14159265358979323846f

typedef float v2f __attribute__((ext_vector_type(2)));
typedef float v8f __attribute__((ext_vector_type(8)));

// ---------------- complex helpers ----------------
struct cpx { float x, y; };

__device__ __forceinline__ cpx cmul(cpx a, cpx b) {
  cpx r; r.x = a.x * b.x - a.y * b.y; r.y = a.x * b.y + a.y * b.x; return r;
}
__device__ __forceinline__ cpx cadd(cpx a, cpx b) { cpx r; r.x = a.x + b.x; r.y = a.y + b.y; return r; }
__device__ __forceinline__ cpx csel(bool c, cpx a, cpx b) {
  cpx r; r.x = c ? a.x : b.x; r.y = c ? a.y : b.y; return r;
}

struct mat2 { cpx m00, m01, m10, m11; };

__device__ __forceinline__ mat2 mmul(mat2 A, mat2 B) {
  mat2 C;
  C.m00 = cadd(cmul(A.m00, B.m00), cmul(A.m01, B.m10));
  C.m01 = cadd(cmul(A.m00, B.m01), cmul(A.m01, B.m11));
  C.m10 = cadd(cmul(A.m10, B.m00), cmul(A.m11, B.m10));
  C.m11 = cadd(cmul(A.m10, B.m01), cmul(A.m11, B.m11));
  return C;
}
__device__ __forceinline__ mat2 mRX(float t) {
  float c = __cosf(0.5f * t), s = __sinf(0.5f * t);
  mat2 m; m.m00 = {c, 0.f}; m.m01 = {0.f, -s}; m.m10 = {0.f, -s}; m.m11 = {c, 0.f}; return m;
}
__device__ __forceinline__ mat2 mRY(float t) {
  float c = __cosf(0.5f * t), s = __sinf(0.5f * t);
  mat2 m; m.m00 = {c, 0.f}; m.m01 = {-s, 0.f}; m.m10 = {s, 0.f}; m.m11 = {c, 0.f}; return m;
}
__device__ __forceinline__ mat2 mRZ(float t) {
  float c = __cosf(0.5f * t), s = __sinf(0.5f * t);
  mat2 m; m.m00 = {c, -s}; m.m01 = {0.f, 0.f}; m.m10 = {0.f, 0.f}; m.m11 = {c, s}; return m;
}

// ---------------- register-resident statevector primitives (wave32) ----------------
// amp index i[7:0]: lane = i[7:3], slot = i[2:0]; wire w <-> bit (7-w)

template <int D>
__device__ __forceinline__ void gate_lane(cpx amp[8], cpx u00, cpx u01, cpx u10, cpx u11, int lane) {
  bool hi = (lane & D) != 0;
  cpx cs = csel(hi, u11, u00);
  cpx co = csel(hi, u10, u01);
#pragma unroll
  for (int s = 0; s < 8; ++s) {
    cpx p;
    p.x = __shfl_xor(amp[s].x, D, 32);
    p.y = __shfl_xor(amp[s].y, D, 32);
    amp[s] = cadd(cmul(cs, amp[s]), cmul(co, p));
  }
}

template <int M>
__device__ __forceinline__ void gate_slot(cpx amp[8], cpx u00, cpx u01, cpx u10, cpx u11) {
#pragma unroll
  for (int s = 0; s < 8; ++s) {
    if (!(s & M)) {
      cpx a0 = amp[s], a1 = amp[s | M];
      amp[s]     = cadd(cmul(u00, a0), cmul(u01, a1));
      amp[s | M] = cadd(cmul(u10, a0), cmul(u11, a1));
    }
  }
}

template <int DC, int DT>
__device__ __forceinline__ void cnot_ll(cpx amp[8], int lane) {
  bool c = (lane & DC) != 0;
#pragma unroll
  for (int s = 0; s < 8; ++s) {
    cpx p;
    p.x = __shfl_xor(amp[s].x, DT, 32);
    p.y = __shfl_xor(amp[s].y, DT, 32);
    amp[s] = csel(c, p, amp[s]);
  }
}

template <int DC, int MT>
__device__ __forceinline__ void cnot_ls(cpx amp[8], int lane) {
  bool c = (lane & DC) != 0;
  cpx t[8];
#pragma unroll
  for (int s = 0; s < 8; ++s) t[s] = csel(c, amp[s ^ MT], amp[s]);
#pragma unroll
  for (int s = 0; s < 8; ++s) amp[s] = t[s];
}

template <int MC, int MT>
__device__ __forceinline__ void cnot_ss(cpx amp[8]) {
#pragma unroll
  for (int s = 0; s < 8; ++s) {
    if ((s & MC) && !(s & MT)) { cpx t = amp[s]; amp[s] = amp[s | MT]; amp[s | MT] = t; }
  }
}

// ---------------- kernel 1: build real-ified variational unitary A = [[Ur,-Ui],[Ui,Ur]] ----------------
// A is 512x512 f32 row-major in d_ws. 256 columns of U, one per wave (32 blocks x 8 waves).
__global__ void __launch_bounds__(256) build_U_kernel(const float* __restrict__ W, float* __restrict__ A) {
  __shared__ float gl[4 * 8 * 8];  // fused gates: [layer][wire][8 floats]
  int tid = threadIdx.x;
  if (tid < 32) {
    int l = tid >> 3, wi = tid & 7;
    float a0 = W[(4 * l + 0) * 8 + wi];
    float a1 = W[(4 * l + 1) * 8 + wi];
    float a2 = W[(4 * l + 2) * 8 + wi];
    float a3 = W[(4 * l + 3) * 8 + wi];
    // applied order RX(a0), RY(a1), RZ(a2), RX(a3)  ->  G = RX3*RZ2*RY1*RX0
    mat2 g = mmul(mRY(a1), mRX(a0));
    g = mmul(mRZ(a2), g);
    g = mmul(mRX(a3), g);
    float* o = &gl[tid * 8];
    o[0] = g.m00.x; o[1] = g.m00.y; o[2] = g.m01.x; o[3] = g.m01.y;
    o[4] = g.m10.x; o[5] = g.m10.y; o[6] = g.m11.x; o[7] = g.m11.y;
  }
  __syncthreads();

  int lane = tid & 31, wid = tid >> 5;
  int j = blockIdx.x * 8 + wid;  // basis column 0..255

  cpx amp[8];
#pragma unroll
  for (int s = 0; s < 8; ++s) {
    amp[s].x = ((lane == (j >> 3)) && (s == (j & 7))) ? 1.0f : 0.0f;
    amp[s].y = 0.0f;
  }

#pragma unroll
  for (int l = 0; l < 4; ++l) {
    cpx u00, u01, u10, u11;
#define LOADG(w)                                                         \
  {                                                                      \
    const float* gp = &gl[(l * 8 + (w)) * 8];                            \
    u00.x = gp[0]; u00.y = gp[1]; u01.x = gp[2]; u01.y = gp[3];          \
    u10.x = gp[4]; u10.y = gp[5]; u11.x = gp[6]; u11.y = gp[7];          \
  }
    LOADG(0); gate_lane<16>(amp, u00, u01, u10, u11, lane);  // wire0 -> bit7
    LOADG(1); gate_lane<8>(amp, u00, u01, u10, u11, lane);
    LOADG(2); gate_lane<4>(amp, u00, u01, u10, u11, lane);
    LOADG(3); gate_lane<2>(amp, u00, u01, u10, u11, lane);
    LOADG(4); gate_lane<1>(amp, u00, u01, u10, u11, lane);
    LOADG(5); gate_slot<4>(amp, u00, u01, u10, u11);
    LOADG(6); gate_slot<2>(amp, u00, u01, u10, u11);
    LOADG(7); gate_slot<1>(amp, u00, u01, u10, u11);
#undef LOADG
    // CNOT ring: (0,1)(1,2)(2,3)(3,4)(4,5)(5,6)(6,7)(0,7)(1,6)
    cnot_ll<16, 8>(amp, lane);
    cnot_ll<8, 4>(amp, lane);
    cnot_ll<4, 2>(amp, lane);
    cnot_ll<2, 1>(amp, lane);
    cnot_ls<1, 4>(amp, lane);   // ctrl wire4 (lane bit0), tgt wire5 (slot bit2)
    cnot_ss<4, 2>(amp);         // (5,6)
    cnot_ss<2, 1>(amp);         // (6,7)
    cnot_ls<16, 1>(amp, lane);  // (0,7)
    cnot_ls<8, 2>(amp, lane);   // (1,6)
  }

  // write column j into real-ified A (row-major 512x512)
#pragma unroll
  for (int s = 0; s < 8; ++s) {
    int i = (lane << 3) | s;
    A[i * 512 + j]               = amp[s].x;
    A[(i + 256) * 512 + j]       = amp[s].y;
    A[i * 512 + j + 256]         = -amp[s].y;
    A[(i + 256) * 512 + j + 256] = amp[s].x;
  }
}

// ---------------- kernel 2: encoded-state build + WMMA GEMM + expectations ----------------
// One workgroup (8 waves) per 32 batch columns; each wave owns paired Re/Im row tiles so
// |amp|^2 and the Z-expectations are formed from accumulators in-register (no Y spill).
__global__ void __launch_bounds__(256) gemm_expect_kernel(const float* __restrict__ x,
                                                          const float* __restrict__ A,
                                                          float* __restrict__ out) {
  __shared__ float smem[512 * 32];          // B = [Sr;Si], 512 rows x 32 cols (64 KB)
  __shared__ float part[8 * 2 * 16 * 8];    // [wave][ntile][col16][wire] partials (8 KB)

  int tid = threadIdx.x;
  int n0 = blockIdx.x * 32;
  int lane = tid & 31, wid = tid >> 5;

  // ---- build encoded product states for 32 batch columns: column = lane, rows g*32.. ----
  {
    int n = lane;            // column 0..31
    int g = wid;             // row group: amps g*32 .. g*32+31
    int batch = n0 + n;
    cpx v0[8], v1[8];
#pragma unroll
    for (int w = 0; w < 8; ++w) {
      float xv = x[batch * 8 + w];
      float ha = 0.5f * PI_F * xv;   // RY half-angle
      float hb = 0.25f * PI_F * xv;  // RZ half-angle
      float cy = __cosf(ha), sy = __sinf(ha);
      float cz = __cosf(hb), sz = __sinf(hb);
      v0[w].x = cy * cz; v0[w].y = -cy * sz;
      v1[w].x = sy * cz; v1[w].y = sy * sz;
    }
    // i = g*32 + b*16 + lo ; hi nibble = {g[2:0], b} <-> wires 0..3, lo nibble <-> wires 4..7
    cpx thg = csel(g & 4, v1[0], v0[0]);
    thg = cmul(thg, csel(g & 2, v1[1], v0[1]));
    thg = cmul(thg, csel(g & 1, v1[2], v0[2]));
    cpx th0 = cmul(thg, v0[3]);
    cpx th1 = cmul(thg, v1[3]);
    cpx tl[16];
#pragma unroll
    for (int lo = 0; lo < 16; ++lo) {
      cpx p = (lo & 8) ? v1[4] : v0[4];
      p = cmul(p, (lo & 4) ? v1[5] : v0[5]);
      p = cmul(p, (lo & 2) ? v1[6] : v0[6]);
      p = cmul(p, (lo & 1) ? v1[7] : v0[7]);
      tl[lo] = p;
    }
#pragma unroll
    for (int b = 0; b < 2; ++b) {
#pragma unroll
      for (int lo = 0; lo < 16; ++lo) {
        int i = g * 32 + b * 16 + lo;
        cpx a = cmul(b ? th1 : th0, tl[lo]);
        smem[i * 32 + n]         = a.x;  // Sr
        smem[(i + 256) * 32 + n] = a.y;  // Si
      }
    }
  }
  __syncthreads();

  // ---- GEMM: Y(512x32) = A(512x512) * B(512x32) via v_wmma_f32_16x16x4_f32 ----
  // Wave wid owns rows [32w,32w+32) (Re) and [256+32w, 256+32w+32) (Im): tiles h=0..3.
  int col = lane & 15;
  int khalf = (lane >> 4) << 1;  // lanes 16-31 hold K=2,3 halves
  const v8f vzero = {0.f, 0.f, 0.f, 0.f, 0.f, 0.f, 0.f, 0.f};
  v8f acc[4][2];
#pragma unroll
  for (int h = 0; h < 4; ++h) {
    acc[h][0] = vzero;
    acc[h][1] = vzero;
  }

  const float* Ab[4];
  Ab[0] = A + (wid * 32 + col) * 512;            // Re tile 0
  Ab[1] = A + (wid * 32 + 16 + col) * 512;       // Re tile 1
  Ab[2] = A + (256 + wid * 32 + col) * 512;      // Im tile 0
  Ab[3] = A + (256 + wid * 32 + 16 + col) * 512; // Im tile 1

  for (int k0 = 0; k0 < 512; k0 += 4) {
    int k = k0 + khalf;
    v2f bf0, bf1;
    bf0.x = smem[k * 32 + col];
    bf0.y = smem[(k + 1) * 32 + col];
    bf1.x = smem[k * 32 + 16 + col];
    bf1.y = smem[(k + 1) * 32 + 16 + col];
#pragma unroll
    for (int h = 0; h < 4; ++h) {
      const float* ap = Ab[h] + k;
      v2f af;
      af.x = ap[0];
      af.y = ap[1];
      acc[h][0] = __builtin_amdgcn_wmma_f32_16x16x4_f32(false, af, false, bf0,
                                                        (short)0, acc[h][0], false, false);
      acc[h][1] = __builtin_amdgcn_wmma_f32_16x16x4_f32(false, af, false, bf1,
                                                        (short)0, acc[h][1], false, false);
    }
  }

  // ---- expectations, fully in-register: i = {wid[2:0], h, half, r[2:0]} ----
  int half = lane >> 4;
  float aw[2][8];
#pragma unroll
  for (int nt = 0; nt < 2; ++nt)
#pragma unroll
    for (int q = 0; q < 8; ++q) aw[nt][q] = 0.f;

#pragma unroll
  for (int h = 0; h < 2; ++h) {
#pragma unroll
    for (int r = 0; r < 8; ++r) {
#pragma unroll
      for (int nt = 0; nt < 2; ++nt) {
        float yr = acc[h][nt][r];
        float yi = acc[h + 2][nt][r];
        float p = yr * yr + yi * yi;
        aw[nt][0] += (wid & 4) ? -p : p;   // wire0 <- bit7 = wid[2]
        aw[nt][1] += (wid & 2) ? -p : p;   // wire1 <- bit6 = wid[1]
        aw[nt][2] += (wid & 1) ? -p : p;   // wire2 <- bit5 = wid[0]
        aw[nt][3] += (h) ? -p : p;         // wire3 <- bit4 = h        (folds)
        aw[nt][4] += (half) ? -p : p;      // wire4 <- bit3 = lanehalf
        aw[nt][5] += (r & 4) ? -p : p;     // wire5 <- bit2            (folds)
        aw[nt][6] += (r & 2) ? -p : p;     // wire6 <- bit1            (folds)
        aw[nt][7] += (r & 1) ? -p : p;     // wire7 <- bit0            (folds)
      }
    }
  }
  // combine lane halves (rows differing in bit3 share a column)
#pragma unroll
  for (int nt = 0; nt < 2; ++nt)
#pragma unroll
    for (int q = 0; q < 8; ++q) aw[nt][q] += __shfl_xor(aw[nt][q], 16, 32);

  if (lane < 16) {
#pragma unroll
    for (int nt = 0; nt < 2; ++nt)
#pragma unroll
      for (int q = 0; q < 8; ++q)
        part[((wid * 2 + nt) * 16 + col) * 8 + q] = aw[nt][q];
  }
  __syncthreads();

  // ---- cross-wave combine + store (B,8) ----
  {
    int n = tid >> 3;   // 0..31
    int q = tid & 7;    // wire
    float s = 0.f;
#pragma unroll
    for (int wv = 0; wv < 8; ++wv)
      s += part[((wv * 2 + (n >> 4)) * 16 + (n & 15)) * 8 + q];
    out[(n0 + n) * 8 + q] = s;
  }
}

// ---------------- launch ----------------
extern "C" void kernel_launch(void* const* d_in, const int* in_sizes, int n_in,
                              void* d_out, int out_size, void* d_ws, size_t ws_size,
                              hipStream_t stream) {
  const float* x = (const float*)d_in[0];   // (8192, 8) f32
  const float* W = (const float*)d_in[1];   // (16, 8) f32
  float* out = (float*)d_out;               // (8192, 8) f32
  float* A = (float*)d_ws;                  // 512*512 f32 = 1 MB scratch

  build_U_kernel<<<32, 256, 0, stream>>>(W, A);
  gemm_expect_kernel<<<256, 256, 0, stream>>>(x, A, out);
  (void)in_sizes; (void)n_in; (void)out_size; (void)ws_size;
}